// GATv2_1382979470034
// MI455X (gfx1250) — compile-verified
//
#include <hip/hip_runtime.h>
#include <hip/hip_bf16.h>
#include <math.h>

typedef float v2f __attribute__((ext_vector_type(2)));
typedef float v8f __attribute__((ext_vector_type(8)));
typedef int   v4i __attribute__((ext_vector_type(4)));

#define NN     50000          // divisible by 16 -> no row guards in GEMMs
#define EE     1600000
#define EA     (EE + NN)      // edges + self loops
#define FEATD  512
#define D1     64             // HEADS*HID
#define HEADSN 8
#define HIDN   8
#define NCLS   7
#define NEGS   0.2f

#define APAD   4              // LDS row pad (floats): 16B-aligned rows,
                              // bank = (4*r + k) % 64 distinct for r=0..15

#if __has_builtin(__builtin_amdgcn_global_load_async_to_lds_b128)
#define HAVE_ASYNC_LDS 1
typedef __attribute__((address_space(1))) void AS1void;
typedef __attribute__((address_space(3))) void AS3void;
typedef __attribute__((address_space(1))) v4i* as1_v4i_p;
typedef __attribute__((address_space(3))) v4i* as3_v4i_p;
#endif

static __device__ __forceinline__ void async_fence_then_barrier() {
#if defined(HAVE_ASYNC_LDS)
#if __has_builtin(__builtin_amdgcn_s_wait_asynccnt)
  __builtin_amdgcn_s_wait_asynccnt(0);
#else
  asm volatile("s_wait_asynccnt 0x0" ::: "memory");
#endif
#endif
  __syncthreads();
}

// ---------------------------------------------------------------------------
// float atomic max via ordered-int trick (lowers to global_atomic_max/min)
// ---------------------------------------------------------------------------
static __device__ __forceinline__ void atomicMaxF(float* addr, float v) {
  if (v >= 0.0f)
    atomicMax((int*)addr, __float_as_int(v));
  else
    atomicMin((unsigned int*)addr, __float_as_uint(v));
}

// ---------------------------------------------------------------------------
// init: emax = -inf, accumulators = 0
// ---------------------------------------------------------------------------
__global__ __launch_bounds__(256) void k_init(float* __restrict__ emax1,
                                              float* __restrict__ den1,
                                              float* __restrict__ out1,
                                              float* __restrict__ emax2,
                                              float* __restrict__ den2,
                                              float* __restrict__ out2) {
  int t = blockIdx.x * 256 + threadIdx.x;
  if (t >= NN * 90) return;
  if (t < NN * 8)            { emax1[t] = -INFINITY; return; }
  if (t < NN * 16)           { den1[t - NN * 8] = 0.0f; return; }
  if (t < NN * 80)           { out1[t - NN * 16] = 0.0f; return; }
  if (t < NN * 81)           { emax2[t - NN * 80] = -INFINITY; return; }
  if (t < NN * 82)           { den2[t - NN * 81] = 0.0f; return; }
  out2[t - NN * 82] = 0.0f;
}

// ---------------------------------------------------------------------------
// Layer-1 projections: XL = x@Wl1+bl1, XR = x@Wr1+br1   (WMMA f32 16x16x4)
// 256 threads = 8 waves: wave = {L|R} x 4 column tiles of 16, same 16 rows.
// A tile staged to LDS via async global->LDS engine (ASYNCcnt).
// ---------------------------------------------------------------------------
__global__ __launch_bounds__(256) void k_gemm1(const float* __restrict__ x,
                                               const float* __restrict__ Wl,
                                               const float* __restrict__ bl,
                                               const float* __restrict__ Wr,
                                               const float* __restrict__ br,
                                               float* __restrict__ XL,
                                               float* __restrict__ XR) {
  __shared__ float tA[16][FEATD + APAD];
  const int row0 = blockIdx.x * 16;
  const int tid  = threadIdx.x;

#if defined(HAVE_ASYNC_LDS)
  for (int i = tid; i < 16 * (FEATD / 4); i += 256) {
    const int r  = i >> 7;                      // FEATD/4 == 128
    const int c4 = i & 127;
    const float* gp = x + (size_t)(row0 + r) * FEATD + c4 * 4;
    __builtin_amdgcn_global_load_async_to_lds_b128(
        (as1_v4i_p)(AS1void*)(void*)const_cast<float*>(gp),
        (as3_v4i_p)(AS3void*)(void*)&tA[r][c4 * 4], 0, 0);
  }
#else
  for (int i = tid; i < 16 * (FEATD / 4); i += 256) {
    const int r  = i >> 7;
    const int c4 = i & 127;
    const float4 v = ((const float4*)(x + (size_t)(row0 + r) * FEATD))[c4];
    tA[r][c4 * 4 + 0] = v.x; tA[r][c4 * 4 + 1] = v.y;
    tA[r][c4 * 4 + 2] = v.z; tA[r][c4 * 4 + 3] = v.w;
  }
#endif
  async_fence_then_barrier();

  const int wave  = tid >> 5;
  const int lane  = tid & 31;
  const int which = wave & 1;                   // 0 = left, 1 = right proj
  const int col0  = (wave >> 1) << 4;           // 0,16,32,48
  const float* W  = which ? Wr : Wl;
  const float* bb = which ? br : bl;
  float*       O  = which ? XR : XL;

  const int r     = lane & 15;                  // M row within tile
  const int klane = (lane >> 4) << 1;           // K sub-offset: 0 or 2
  const int n     = lane & 15;                  // N column within tile

  v8f acc = {0.f, 0.f, 0.f, 0.f, 0.f, 0.f, 0.f, 0.f};
  for (int k = 0; k < FEATD; k += 4) {
    const int kk = k + klane;
    v2f a, b;
    a[0] = tA[r][kk];
    a[1] = tA[r][kk + 1];
    b[0] = W[(size_t)kk * D1 + col0 + n];
    b[1] = W[(size_t)(kk + 1) * D1 + col0 + n];
    if (k + 8 < FEATD)                          // weights are L2-resident
      __builtin_prefetch(&W[(size_t)(kk + 8) * D1 + col0 + n], 0, 1);
    acc = __builtin_amdgcn_wmma_f32_16x16x4_f32(false, a, false, b,
                                                (short)0, acc, false, false);
  }

  const float bias = bb[col0 + n];
  const int   mhi  = (lane >> 4) << 3;          // 0 or 8
#pragma unroll
  for (int v = 0; v < 8; ++v) {
    const int row = row0 + mhi + v;
    O[(size_t)row * D1 + col0 + n] = acc[v] + bias;
  }
}

// ---------------------------------------------------------------------------
// Layer-1 edge scores: 16 lanes per edge, one float4 (4 channels) per lane.
// e[h] = att1[h] . leaky_relu(XL[s] + XR[d]); also segment-max into emax.
// ---------------------------------------------------------------------------
__global__ __launch_bounds__(256) void k_escore1(const int* __restrict__ ei,
                                                 const float* __restrict__ XL,
                                                 const float* __restrict__ XR,
                                                 const float* __restrict__ att,
                                                 float* __restrict__ esc,
                                                 float* __restrict__ emax) {
  const int g = blockIdx.x * 16 + (threadIdx.x >> 4);
  if (g >= EA) return;
  const int sub = threadIdx.x & 15;
  int s, d;
  if (g < EE) { s = ei[g]; d = ei[EE + g]; } else { s = g - EE; d = s; }

  const float4 xl = ((const float4*)(XL + (size_t)s * D1))[sub];
  const float4 xr = ((const float4*)(XR + (size_t)d * D1))[sub];
  const int h = sub >> 1;
  const float* a = att + h * HIDN + (sub & 1) * 4;

  float m, p = 0.f;
  m = xl.x + xr.x; m = m > 0.f ? m : NEGS * m; p += m * a[0];
  m = xl.y + xr.y; m = m > 0.f ? m : NEGS * m; p += m * a[1];
  m = xl.z + xr.z; m = m > 0.f ? m : NEGS * m; p += m * a[2];
  m = xl.w + xr.w; m = m > 0.f ? m : NEGS * m; p += m * a[3];
  p += __shfl_xor(p, 1, 32);                    // pair-reduce (8 ch / head)

  if ((sub & 1) == 0) {
    esc[(size_t)g * HEADSN + h] = p;
    atomicMaxF(&emax[(size_t)d * HEADSN + h], p);
  }
}

// ---------------------------------------------------------------------------
// Layer-1 softmax denominator; overwrite score with exp(e - emax[d]).
// ---------------------------------------------------------------------------
__global__ __launch_bounds__(256) void k_esoft1(const int* __restrict__ ei,
                                                float* __restrict__ esc,
                                                const float* __restrict__ emax,
                                                float* __restrict__ den) {
  const int t = blockIdx.x * 256 + threadIdx.x;
  if (t >= EA * HEADSN) return;
  const int g = t >> 3, h = t & 7;
  const int d = (g < EE) ? ei[EE + g] : (g - EE);
  const float ex = __expf(esc[t] - emax[(size_t)d * HEADSN + h]);
  esc[t] = ex;
  atomicAdd(&den[(size_t)d * HEADSN + h], ex);
}

// ---------------------------------------------------------------------------
// Layer-1 aggregation: out[d] += alpha * XL[s]  (16 lanes x float4 per edge)
// ---------------------------------------------------------------------------
__global__ __launch_bounds__(256) void k_agg1(const int* __restrict__ ei,
                                              const float* __restrict__ XL,
                                              const float* __restrict__ esc,
                                              const float* __restrict__ den,
                                              float* __restrict__ out) {
  const int g = blockIdx.x * 16 + (threadIdx.x >> 4);
  if (g >= EA) return;
  const int sub = threadIdx.x & 15;
  int s, d;
  if (g < EE) { s = ei[g]; d = ei[EE + g]; } else { s = g - EE; d = s; }

  const int   h     = sub >> 1;
  const float alpha = esc[(size_t)g * HEADSN + h] /
                      den[(size_t)d * HEADSN + h];
  const float4 xl = ((const float4*)(XL + (size_t)s * D1))[sub];
  float* o = out + (size_t)d * D1 + sub * 4;
  atomicAdd(o + 0, alpha * xl.x);
  atomicAdd(o + 1, alpha * xl.y);
  atomicAdd(o + 2, alpha * xl.z);
  atomicAdd(o + 3, alpha * xl.w);
}

// ---------------------------------------------------------------------------
// H = elu(out1 + bias1)
// ---------------------------------------------------------------------------
__global__ __launch_bounds__(256) void k_elu(const float* __restrict__ out1,
                                             const float* __restrict__ bias1,
                                             float* __restrict__ H) {
  const int t = blockIdx.x * 256 + threadIdx.x;
  if (t >= NN * D1) return;
  const float v = out1[t] + bias1[t & 63];
  H[t] = v > 0.f ? v : (__expf(v) - 1.f);
}

// ---------------------------------------------------------------------------
// Layer-2 projections: XL2/XR2 [N,8] (7 classes + 1 zero pad), WMMA f32.
// 2 waves per 16-row tile (left/right).
// ---------------------------------------------------------------------------
__global__ __launch_bounds__(64) void k_gemm2(const float* __restrict__ Hm,
                                              const float* __restrict__ Wl,
                                              const float* __restrict__ bl,
                                              const float* __restrict__ Wr,
                                              const float* __restrict__ br,
                                              float* __restrict__ XL2,
                                              float* __restrict__ XR2) {
  __shared__ float tA[16][D1 + APAD];
  const int row0 = blockIdx.x * 16;
  const int tid  = threadIdx.x;

#if defined(HAVE_ASYNC_LDS)
  for (int i = tid; i < 16 * (D1 / 4); i += 64) {
    const int r  = i >> 4;                      // D1/4 == 16
    const int c4 = i & 15;
    const float* gp = Hm + (size_t)(row0 + r) * D1 + c4 * 4;
    __builtin_amdgcn_global_load_async_to_lds_b128(
        (as1_v4i_p)(AS1void*)(void*)const_cast<float*>(gp),
        (as3_v4i_p)(AS3void*)(void*)&tA[r][c4 * 4], 0, 0);
  }
#else
  for (int i = tid; i < 16 * (D1 / 4); i += 64) {
    const int r  = i >> 4;
    const int c4 = i & 15;
    const float4 v = ((const float4*)(Hm + (size_t)(row0 + r) * D1))[c4];
    tA[r][c4 * 4 + 0] = v.x; tA[r][c4 * 4 + 1] = v.y;
    tA[r][c4 * 4 + 2] = v.z; tA[r][c4 * 4 + 3] = v.w;
  }
#endif
  async_fence_then_barrier();

  const int wave = tid >> 5;                    // 0 = left, 1 = right
  const int lane = tid & 31;
  const float* W  = wave ? Wr : Wl;
  const float* bb = wave ? br : bl;
  float*       O  = wave ? XR2 : XL2;

  const int r     = lane & 15;
  const int klane = (lane >> 4) << 1;
  const int n     = lane & 15;

  v8f acc = {0.f, 0.f, 0.f, 0.f, 0.f, 0.f, 0.f, 0.f};
  for (int k = 0; k < D1; k += 4) {
    const int kk = k + klane;
    v2f a, b;
    a[0] = tA[r][kk];
    a[1] = tA[r][kk + 1];
    b[0] = (n < NCLS) ? W[kk * NCLS + n] : 0.f;
    b[1] = (n < NCLS) ? W[(kk + 1) * NCLS + n] : 0.f;
    acc = __builtin_amdgcn_wmma_f32_16x16x4_f32(false, a, false, b,
                                                (short)0, acc, false, false);
  }

  const float bias = (n < NCLS) ? bb[n] : 0.f;
  const int   mhi  = (lane >> 4) << 3;
#pragma unroll
  for (int v = 0; v < 8; ++v) {
    const int row = row0 + mhi + v;
    if (n < 8)
      O[(size_t)row * 8 + n] = (n < NCLS) ? (acc[v] + bias) : 0.f;
  }
}

// ---------------------------------------------------------------------------
// Layer-2 edge scores: 8 lanes per edge (c = 0..7, c==7 is zero pad).
// ---------------------------------------------------------------------------
__global__ __launch_bounds__(256) void k_escore2(const int* __restrict__ ei,
                                                 const float* __restrict__ XL2,
                                                 const float* __restrict__ XR2,
                                                 const float* __restrict__ att2,
                                                 float* __restrict__ esc2,
                                                 float* __restrict__ emax2) {
  const int g = blockIdx.x * 32 + (threadIdx.x >> 3);
  if (g >= EA) return;
  const int c = threadIdx.x & 7;
  int s, d;
  if (g < EE) { s = ei[g]; d = ei[EE + g]; } else { s = g - EE; d = s; }

  const float xl = XL2[(size_t)s * 8 + c];
  const float xr = XR2[(size_t)d * 8 + c];
  float m = xl + xr; m = m > 0.f ? m : NEGS * m;
  float p = (c < NCLS) ? m * att2[c] : 0.f;
  p += __shfl_xor(p, 1, 32);
  p += __shfl_xor(p, 2, 32);
  p += __shfl_xor(p, 4, 32);
  if (c == 0) {
    esc2[g] = p;
    atomicMaxF(&emax2[d], p);
  }
}

__global__ __launch_bounds__(256) void k_esoft2(const int* __restrict__ ei,
                                                float* __restrict__ esc2,
                                                const float* __restrict__ emax2,
                                                float* __restrict__ den2) {
  const int g = blockIdx.x * 256 + threadIdx.x;
  if (g >= EA) return;
  const int d = (g < EE) ? ei[EE + g] : (g - EE);
  const float ex = __expf(esc2[g] - emax2[d]);
  esc2[g] = ex;
  atomicAdd(&den2[d], ex);
}

__global__ __launch_bounds__(256) void k_agg2(const int* __restrict__ ei,
                                              const float* __restrict__ XL2,
                                              const float* __restrict__ esc2,
                                              const float* __restrict__ den2,
                                              float* __restrict__ out2) {
  const int g = blockIdx.x * 32 + (threadIdx.x >> 3);
  if (g >= EA) return;
  const int c = threadIdx.x & 7;
  int s, d;
  if (g < EE) { s = ei[g]; d = ei[EE + g]; } else { s = g - EE; d = s; }
  const float alpha = esc2[g] / den2[d];
  if (c < NCLS)
    atomicAdd(&out2[(size_t)d * 8 + c], alpha * XL2[(size_t)s * 8 + c]);
}

// ---------------------------------------------------------------------------
// log_softmax(out2 + bias2) -> d_out [N, 7]
// ---------------------------------------------------------------------------
__global__ __launch_bounds__(256) void k_final(const float* __restrict__ out2,
                                               const float* __restrict__ bias2,
                                               float* __restrict__ out) {
  const int nId = blockIdx.x * 32 + (threadIdx.x >> 3);
  if (nId >= NN) return;
  const int c = threadIdx.x & 7;
  const float v = (c < NCLS) ? out2[(size_t)nId * 8 + c] + bias2[c]
                             : -INFINITY;
  float mx = v;
  mx = fmaxf(mx, __shfl_xor(mx, 1, 32));
  mx = fmaxf(mx, __shfl_xor(mx, 2, 32));
  mx = fmaxf(mx, __shfl_xor(mx, 4, 32));
  float ex = (c < NCLS) ? __expf(v - mx) : 0.f;
  float sum = ex;
  sum += __shfl_xor(sum, 1, 32);
  sum += __shfl_xor(sum, 2, 32);
  sum += __shfl_xor(sum, 4, 32);
  if (c < NCLS) out[(size_t)nId * NCLS + c] = v - mx - __logf(sum);
}

// ---------------------------------------------------------------------------
extern "C" void kernel_launch(void* const* d_in, const int* in_sizes, int n_in,
                              void* d_out, int out_size, void* d_ws,
                              size_t ws_size, hipStream_t stream) {
  const float* x     = (const float*)d_in[0];
  const int*   ei    = (const int*)d_in[1];     // [2,E]: row0 = src, row1 = dst
  const float* Wl1   = (const float*)d_in[2];
  const float* bl1   = (const float*)d_in[3];
  const float* Wr1   = (const float*)d_in[4];
  const float* br1   = (const float*)d_in[5];
  const float* att1  = (const float*)d_in[6];
  const float* bias1 = (const float*)d_in[7];
  const float* Wl2   = (const float*)d_in[8];
  const float* bl2   = (const float*)d_in[9];
  const float* Wr2   = (const float*)d_in[10];
  const float* br2   = (const float*)d_in[11];
  const float* att2  = (const float*)d_in[12];
  const float* bias2 = (const float*)d_in[13];

  float* ws = (float*)d_ws;
  size_t off = 0;
  float* XL1   = ws + off; off += (size_t)NN * D1;
  float* XR1   = ws + off; off += (size_t)NN * D1;   // reused as H after agg1
  float* ESC1  = ws + off; off += (size_t)EA * HEADSN;
  float* EMAX1 = ws + off; off += (size_t)NN * HEADSN;
  float* DEN1  = ws + off; off += (size_t)NN * HEADSN;
  float* OUT1  = ws + off; off += (size_t)NN * D1;
  float* XL2   = ws + off; off += (size_t)NN * 8;
  float* XR2   = ws + off; off += (size_t)NN * 8;
  float* ESC2  = ws + off; off += (size_t)EA;
  float* EMAX2 = ws + off; off += (size_t)NN;
  float* DEN2  = ws + off; off += (size_t)NN;
  float* OUT2  = ws + off; off += (size_t)NN * 8;
  float* H     = XR1;                                // safe reuse

  k_init   <<<(NN * 90 + 255) / 256, 256, 0, stream>>>(EMAX1, DEN1, OUT1,
                                                       EMAX2, DEN2, OUT2);
  k_gemm1  <<<NN / 16, 256, 0, stream>>>(x, Wl1, bl1, Wr1, br1, XL1, XR1);
  k_escore1<<<(EA + 15) / 16, 256, 0, stream>>>(ei, XL1, XR1, att1, ESC1, EMAX1);
  k_esoft1 <<<(EA * HEADSN + 255) / 256, 256, 0, stream>>>(ei, ESC1, EMAX1, DEN1);
  k_agg1   <<<(EA + 15) / 16, 256, 0, stream>>>(ei, XL1, ESC1, DEN1, OUT1);
  k_elu    <<<(NN * D1 + 255) / 256, 256, 0, stream>>>(OUT1, bias1, H);
  k_gemm2  <<<NN / 16, 64, 0, stream>>>(H, Wl2, bl2, Wr2, br2, XL2, XR2);
  k_escore2<<<(EA + 31) / 32, 256, 0, stream>>>(ei, XL2, XR2, att2, ESC2, EMAX2);
  k_esoft2 <<<(EA + 255) / 256, 256, 0, stream>>>(ei, ESC2, EMAX2, DEN2);
  k_agg2   <<<(EA + 31) / 32, 256, 0, stream>>>(ei, XL2, ESC2, DEN2, OUT2);
  k_final  <<<(NN + 31) / 32, 256, 0, stream>>>(OUT2, bias2, (float*)d_out);
}